// Attention_86182813761883
// MI455X (gfx1250) — compile-verified
//
#include <hip/hip_runtime.h>

#define N_HEADS    32
#define N_KV_HEADS 8
#define HEAD_DIM   128
#define DIM        4096
#define MAX_BS     32
#define MAX_SEQ    2048

typedef __attribute__((ext_vector_type(2))) float v2f;
typedef __attribute__((ext_vector_type(8))) float v8f;

// ---------------------------------------------------------------------------
// D = A(16x4) x B(4x16) + C, fp32 WMMA.
// A: lanes 0-15 hold rows M=0..15 for K={0,1}; lanes 16-31 same rows, K={2,3}.
// B: lanes 0-15 hold cols N=0..15 of rows K={0,1}; lanes 16-31 rows K={2,3}.
// C/D: VGPR j, lanes 0-15 -> M=j; lanes 16-31 -> M=j+8; N = lane&15.
// ---------------------------------------------------------------------------
__device__ __forceinline__ v8f wmma_f32x4(v2f a, v2f b, v8f c) {
    return __builtin_amdgcn_wmma_f32_16x16x4_f32(
        /*neg_a=*/false, a, /*neg_b=*/false, b,
        /*c_mod=*/(short)0, c, /*reuse_a=*/false, /*reuse_b=*/false);
}

// ---------------------------------------------------------------------------
// C(32 x N) = X(32 x 4096) @ W(4096 x N), row-major.
// One wave per 16-col tile; both 16-row M tiles per wave share the B loads,
// so every W element is read from HBM exactly once (bandwidth-optimal).
// grid = N/64 blocks of 128 threads; EXEC stays all-1s as WMMA requires.
// ---------------------------------------------------------------------------
__global__ void gemm32_wmma(const float* __restrict__ X,
                            const float* __restrict__ W,
                            float* __restrict__ C, int N) {
    const int gwave = (blockIdx.x * blockDim.x + threadIdx.x) >> 5;
    const int lane  = threadIdx.x & 31;
    const int l15   = lane & 15;
    const int koff  = (lane >> 4) << 1;
    const int col   = gwave * 16 + l15;

    const float* x0 = X + (size_t)l15 * DIM;
    const float* x1 = X + (size_t)(16 + l15) * DIM;

    v8f acc0 = {};
    v8f acc1 = {};

#pragma unroll 4
    for (int k = 0; k < DIM; k += 4) {
        const int kk = k + koff;
        v2f a0 = *(const v2f*)(x0 + kk);
        v2f a1 = *(const v2f*)(x1 + kk);
        v2f b;
        b.x = W[(size_t)kk * N + col];
        b.y = W[(size_t)(kk + 1) * N + col];
        acc0 = wmma_f32x4(a0, b, acc0);
        acc1 = wmma_f32x4(a1, b, acc1);
    }

    const int crow = (lane >> 4) * 8;
#pragma unroll
    for (int j = 0; j < 8; ++j) {
        C[(size_t)(crow + j) * N + col]      = acc0[j];
        C[(size_t)(16 + crow + j) * N + col] = acc1[j];
    }
}

// ---------------------------------------------------------------------------
// RoPE on xq (32 x 32 x 128) and xk (32 x 8 x 128); freqs (1 x 64).
// ---------------------------------------------------------------------------
#define Q_PAIRS (MAX_BS * N_HEADS * (HEAD_DIM / 2))
#define K_PAIRS (MAX_BS * N_KV_HEADS * (HEAD_DIM / 2))

__global__ void rope_kernel(float* __restrict__ xq, float* __restrict__ xk,
                            const float* __restrict__ fcos,
                            const float* __restrict__ fsin) {
    const int idx = blockIdx.x * blockDim.x + threadIdx.x;
    if (idx >= Q_PAIRS + K_PAIRS) return;

    float* buf;
    int base, p;
    if (idx < Q_PAIRS) {
        const int b = idx / (N_HEADS * 64);
        const int r = idx % (N_HEADS * 64);
        const int h = r / 64;
        p = r % 64;
        base = b * (N_HEADS * HEAD_DIM) + h * HEAD_DIM + 2 * p;
        buf = xq;
    } else {
        const int i2 = idx - Q_PAIRS;
        const int b = i2 / (N_KV_HEADS * 64);
        const int r = i2 % (N_KV_HEADS * 64);
        const int h = r / 64;
        p = r % 64;
        base = b * (N_KV_HEADS * HEAD_DIM) + h * HEAD_DIM + 2 * p;
        buf = xk;
    }
    const float a  = buf[base];
    const float bb = buf[base + 1];
    const float c  = fcos[p];
    const float d  = fsin[p];
    buf[base]     = a * c - bb * d;
    buf[base + 1] = a * d + bb * c;
}

// ---------------------------------------------------------------------------
// WMMA flash-decode attention, GQA-aware.
// One block (8 waves) per (batch, kv-head); the 4 q-heads sharing that KV
// stream ride as WMMA N-columns (padded to 16), so the 256MB KV cache is read
// once and NO per-key cross-lane reductions are needed:
//   stage 1:  S^T(16keys x 16heads) = K_tile(16x128) @ Q^T      (32 wmma)
//   softmax:  per-lane online max/exp/sum (head == lane&15)
//   stage 2:  O^T(128dims x 16heads) += V^T_tile @ P^T          (32 wmma)
// P^T chunks are built from the score tile with 4 lane-shuffles per K-chunk.
// ~56 bytes of HBM per wave-instruction -> comfortably above the 23.3 TB/s
// roofline requirement (~18 B/inst), unlike a shuffle-reduce formulation.
// t < start_pos reads cache; t == start_pos reads fresh roped xk / raw xv
// (inputs must not be mutated). Per-wave partials merged in LDS.
// ---------------------------------------------------------------------------
#define QSTRIDE 132   // pad 128->132 floats: B-operand ds_load_b64 conflict-free

__global__ void __launch_bounds__(256)
attn_kernel(const float* __restrict__ XQ, const float* __restrict__ XK,
            const float* __restrict__ XV, const float* __restrict__ CK,
            const float* __restrict__ CV, const int* __restrict__ sp_ptr,
            float* __restrict__ AO) {
    const int b     = blockIdx.x >> 3;
    const int hk    = blockIdx.x & 7;
    const int lane  = threadIdx.x & 31;
    const int wave  = threadIdx.x >> 5;
    const int l15   = lane & 15;
    const int lhalf = lane >> 4;           // 0 or 1
    const int koff  = lhalf << 1;          // K offset within WMMA chunk
    const int sp    = *sp_ptr;             // 2047
    const int nkeys = sp + 1;
    const int ntiles = (nkeys + 15) >> 4;
    const float scale = 0.08838834764831845f;   // 1/sqrt(128)

    __shared__ float Qlds[16 * QSTRIDE];
    __shared__ float sm[8][4];
    __shared__ float ss[8][4];
    __shared__ float so[8][4][HEAD_DIM];

    // Q tile: 16 head rows (4 real, 12 zero-padded) x 128 dims
    for (int i = threadIdx.x; i < 16 * QSTRIDE; i += 256) Qlds[i] = 0.f;
    __syncthreads();
    for (int i = threadIdx.x; i < 4 * HEAD_DIM; i += 256) {
        const int r = i >> 7, d = i & 127;
        Qlds[r * QSTRIDE + d] = XQ[(size_t)b * DIM + (hk * 4 + r) * HEAD_DIM + d];
    }
    __syncthreads();

    const float* xk_row = XK + ((size_t)b * N_KV_HEADS + hk) * HEAD_DIM;
    const float* xv_row = XV + ((size_t)b * N_KV_HEADS + hk) * HEAD_DIM;

    float m_run = -1e30f, s_run = 0.f;
    const v8f vzero = {};
    v8f accO[8];
#pragma unroll
    for (int dt = 0; dt < 8; ++dt) accO[dt] = vzero;

    for (int tt = wave; tt < ntiles; tt += 8) {
        const int t0 = tt << 4;

        // ---- stage 1: S^T = K_tile @ Q^T  (A rows = keys, B cols = heads)
        const int trow = t0 + l15;
        const float* kp = (trow < sp)
            ? CK + (((size_t)b * MAX_SEQ + trow) * N_KV_HEADS + hk) * HEAD_DIM
            : xk_row;   // trow == sp: new key; trow > sp: padded (masked below)
        v8f accS = vzero;
#pragma unroll 4
        for (int kk = 0; kk < HEAD_DIM; kk += 4) {
            v2f a  = *(const v2f*)(kp + kk + koff);
            v2f bq = *(const v2f*)(&Qlds[l15 * QSTRIDE + kk + koff]);
            accS = wmma_f32x4(a, bq, accS);
        }

        // ---- online softmax (head = l15; keys j / j+8 across lane halves)
        float sc[8];
        const int kbase = t0 + (lhalf << 3);
#pragma unroll
        for (int j = 0; j < 8; ++j)
            sc[j] = (kbase + j < nkeys) ? accS[j] * scale : -1e30f;
        float tm = sc[0];
#pragma unroll
        for (int j = 1; j < 8; ++j) tm = fmaxf(tm, sc[j]);
        tm = fmaxf(tm, __shfl_xor(tm, 16, 32));       // fold lane halves
        const float mnew = fmaxf(m_run, tm);
        const float corr = __expf(m_run - mnew);
        m_run = mnew;
        float p[8], ps = 0.f;
#pragma unroll
        for (int j = 0; j < 8; ++j) { p[j] = __expf(sc[j] - mnew); ps += p[j]; }
        s_run = s_run * corr + ps;
#pragma unroll
        for (int dt = 0; dt < 8; ++dt)
#pragma unroll
            for (int j = 0; j < 8; ++j) accO[dt][j] *= corr;

        // ---- stage 2: O^T += V^T_tile @ P^T  (A rows = dims, B cols = heads)
#pragma unroll
        for (int kc = 0; kc < 4; ++kc) {
            const int kk = kc << 2;
            // P^T[key k][head l15] lives at lane (l15 + 16*(k>=8)), VGPR k&7
            const float bx0 = __shfl(p[(kk + 0) & 7], l15 + (((kk + 0) >> 3) << 4), 32);
            const float by0 = __shfl(p[(kk + 1) & 7], l15 + (((kk + 1) >> 3) << 4), 32);
            const float bx1 = __shfl(p[(kk + 2) & 7], l15 + (((kk + 2) >> 3) << 4), 32);
            const float by1 = __shfl(p[(kk + 3) & 7], l15 + (((kk + 3) >> 3) << 4), 32);
            v2f bp;
            bp.x = (lhalf == 0) ? bx0 : bx1;
            bp.y = (lhalf == 0) ? by0 : by1;

            const int tA0 = t0 + kk + koff;
            const int tA1 = tA0 + 1;
            const float* vp0 = (tA0 < sp)
                ? CV + (((size_t)b * MAX_SEQ + tA0) * N_KV_HEADS + hk) * HEAD_DIM
                : xv_row;
            const float* vp1 = (tA1 < sp)
                ? CV + (((size_t)b * MAX_SEQ + tA1) * N_KV_HEADS + hk) * HEAD_DIM
                : xv_row;
#pragma unroll
            for (int dt = 0; dt < 8; ++dt) {
                v2f a;
                a.x = vp0[dt * 16 + l15];   // coalesced 64B per lane-half
                a.y = vp1[dt * 16 + l15];
                accO[dt] = wmma_f32x4(a, bp, accO[dt]);
            }
        }
    }

    // fold key-halves of the softmax denominator (m is uniform across halves)
    const float s_tot = s_run + __shfl_xor(s_run, 16, 32);

    if (lane < 16 && l15 < 4) { sm[wave][l15] = m_run; ss[wave][l15] = s_tot; }
#pragma unroll
    for (int dt = 0; dt < 8; ++dt)
#pragma unroll
        for (int j = 0; j < 8; ++j)
            if (l15 < 4)   // head = l15; dim = dt*16 + j + 8*lhalf
                so[wave][l15][dt * 16 + j + (lhalf << 3)] = accO[dt][j];
    __syncthreads();

    // merge 8 wave-partials: 256 threads cover 4 heads x 128 dims in 2 passes
    const int d = threadIdx.x & 127;
    for (int h2 = threadIdx.x >> 7; h2 < 4; h2 += 2) {
        float M = -1e30f;
        for (int w = 0; w < 8; ++w) M = fmaxf(M, sm[w][h2]);
        float S = 0.f, O = 0.f;
        for (int w = 0; w < 8; ++w) {
            const float f = __expf(sm[w][h2] - M);
            S += ss[w][h2] * f;
            O += so[w][h2][d] * f;
        }
        AO[(size_t)b * DIM + (hk * 4 + h2) * HEAD_DIM + d] = O / S;
    }
}

// ---------------------------------------------------------------------------
// Launch.  Inputs: x, wq, wk, wv, wo, cache_k, cache_v, freqs_cos, freqs_sin,
// start_pos.
// ---------------------------------------------------------------------------
extern "C" void kernel_launch(void* const* d_in, const int* in_sizes, int n_in,
                              void* d_out, int out_size, void* d_ws, size_t ws_size,
                              hipStream_t stream) {
    const float* x  = (const float*)d_in[0];
    const float* wq = (const float*)d_in[1];
    const float* wk = (const float*)d_in[2];
    const float* wv = (const float*)d_in[3];
    const float* wo = (const float*)d_in[4];
    const float* ck = (const float*)d_in[5];
    const float* cv = (const float*)d_in[6];
    const float* fc = (const float*)d_in[7];
    const float* fs = (const float*)d_in[8];
    const int*   sp = (const int*)d_in[9];
    float* out = (float*)d_out;

    float* XQ = (float*)d_ws;                        // 32*4096
    float* XK = XQ + MAX_BS * DIM;                   // 32*1024
    float* XV = XK + MAX_BS * N_KV_HEADS * HEAD_DIM; // 32*1024
    float* AO = XV + MAX_BS * N_KV_HEADS * HEAD_DIM; // 32*4096

    gemm32_wmma<<<DIM / 64, 128, 0, stream>>>(x, wq, XQ, DIM);
    gemm32_wmma<<<(N_KV_HEADS * HEAD_DIM) / 64, 128, 0, stream>>>(
        x, wk, XK, N_KV_HEADS * HEAD_DIM);
    gemm32_wmma<<<(N_KV_HEADS * HEAD_DIM) / 64, 128, 0, stream>>>(
        x, wv, XV, N_KV_HEADS * HEAD_DIM);

    rope_kernel<<<(Q_PAIRS + K_PAIRS) / 256, 256, 0, stream>>>(XQ, XK, fc, fs);

    attn_kernel<<<MAX_BS * N_KV_HEADS, 256, 0, stream>>>(XQ, XK, XV, ck, cv, sp, AO);

    gemm32_wmma<<<DIM / 64, 128, 0, stream>>>(AO, wo, out, DIM);
}